// SimpleGNN_17171279250059
// MI455X (gfx1250) — compile-verified
//
#include <hip/hip_runtime.h>

#define N_NODES 100000
#define F_IN 32
#define HDIM 128

typedef float v2f __attribute__((ext_vector_type(2)));
typedef float v8f __attribute__((ext_vector_type(8)));

// ---------------------------------------------------------------- utilities
__global__ void fill_zero_kernel(float* __restrict__ p, long n) {
  long i = (long)blockIdx.x * blockDim.x + threadIdx.x;
  long stride = (long)gridDim.x * blockDim.x;
  for (; i < n; i += stride) p[i] = 0.0f;
}

__global__ void init_out_kernel(float* __restrict__ out,
                                const float* __restrict__ b_lin, int n) {
  int i = blockIdx.x * blockDim.x + threadIdx.x;
  if (i < n) out[i] = b_lin[0];
}

__global__ void degree_kernel(const int* __restrict__ dst,
                              float* __restrict__ deg, long E) {
  long i = (long)blockIdx.x * blockDim.x + threadIdx.x;
  long stride = (long)gridDim.x * blockDim.x;
  for (; i < E; i += stride) atomicAdd(&deg[dst[i]], 1.0f);
}

__global__ void invdeg_kernel(float* __restrict__ deg, int n) {
  int i = blockIdx.x * blockDim.x + threadIdx.x;
  if (i < n) deg[i] = 1.0f / fmaxf(deg[i], 1.0f);
}

// one thread per (edge, feature): accum[dst][f] += feat[src][f]
__global__ void scatter_add_kernel(const float* __restrict__ feat,
                                   const int* __restrict__ src,
                                   const int* __restrict__ dst,
                                   float* __restrict__ accum,
                                   long total, int fshift, int fmask) {
  long i = (long)blockIdx.x * blockDim.x + threadIdx.x;
  long stride = (long)gridDim.x * blockDim.x;
  for (; i < total; i += stride) {
    long e = i >> fshift;
    int f = (int)(i & fmask);
    int s = src[e];
    int d = dst[e];
    atomicAdd(&accum[(((long)d) << fshift) | f], feat[(((long)s) << fshift) | f]);
  }
}

// ---------------------------------------------------------------- layer 1
// h1 = relu( (s1*invdeg) @ W_l1 + x @ W_r1 + b1 )   K = 32, 16 nodes/block
// A tiles staged once per block through LDS (padded +4 dwords: bank-conflict free)
__global__ __launch_bounds__(256) void sage_layer1_kernel(
    const float* __restrict__ x, const float* __restrict__ s1,
    const float* __restrict__ invdeg,
    const float* __restrict__ Wl, const float* __restrict__ Wr,
    const float* __restrict__ b, float* __restrict__ h1) {
  __shared__ float Am[16][F_IN + 4];   // mean rows  (pre-scaled by invdeg)
  __shared__ float Ax[16][F_IN + 4];   // x rows

  const int nodeBase = blockIdx.x * 16;
  {
    const int t = threadIdx.x;
    if (t < 128) {                      // 16 rows * 8 float4 per row
      const int row = t >> 3;
      const int k = (t & 7) * 4;
      const int node = nodeBase + row;
      const float idg = invdeg[node];
      const float4 xs = *(const float4*)(x + (long)node * F_IN + k);
      const float4 ss = *(const float4*)(s1 + (long)node * F_IN + k);
      *(float4*)&Ax[row][k] = xs;
      float4 ms;
      ms.x = ss.x * idg; ms.y = ss.y * idg; ms.z = ss.z * idg; ms.w = ss.w * idg;
      *(float4*)&Am[row][k] = ms;
    }
  }
  __syncthreads();

  const int lane = threadIdx.x & 31;
  const int wave = threadIdx.x >> 5;
  const int colBase = wave * 16;
  const int col = colBase + (lane & 15);
  const int rowA = lane & 15;
  const int khalf = (lane >> 4) * 2;   // 0 or 2

  v8f c = {};
  #pragma unroll
  for (int kk = 0; kk < F_IN; kk += 4) {
    const int k0 = kk + khalf;
    v2f am = *(const v2f*)&Am[rowA][k0];
    v2f ax = *(const v2f*)&Ax[rowA][k0];
    v2f bl, br;
    bl.x = Wl[(long)k0 * HDIM + col]; bl.y = Wl[(long)(k0 + 1) * HDIM + col];
    br.x = Wr[(long)k0 * HDIM + col]; br.y = Wr[(long)(k0 + 1) * HDIM + col];
    c = __builtin_amdgcn_wmma_f32_16x16x4_f32(false, am, false, bl, (short)0, c, false, false);
    c = __builtin_amdgcn_wmma_f32_16x16x4_f32(false, ax, false, br, (short)0, c, false, false);
  }

  const float bias = b[col];
  const int rowOff = (lane >> 4) * 8;  // 0 or 8
  #pragma unroll
  for (int v = 0; v < 8; ++v) {
    float val = fmaxf(c[v] + bias, 0.0f);
    h1[(long)(nodeBase + rowOff + v) * HDIM + col] = val;
  }
}

// ---------------------------------------------------------------- layer 2 (+ fused final projection)
// h2 = (s2*invdeg) @ W_l2 + h1 @ W_r2 + b2 ; out += h2 @ W_lin   K = 128
__global__ __launch_bounds__(256) void sage_layer2_kernel(
    const float* __restrict__ h1, const float* __restrict__ s2,
    const float* __restrict__ invdeg,
    const float* __restrict__ Wl, const float* __restrict__ Wr,
    const float* __restrict__ b, const float* __restrict__ Wlin,
    float* __restrict__ out) {
  __shared__ float Am[16][HDIM + 4];   // mean rows (pre-scaled by invdeg)
  __shared__ float Ah[16][HDIM + 4];   // h1 rows

  const int nodeBase = blockIdx.x * 16;
  {
    const int t = threadIdx.x;
    #pragma unroll
    for (int j = 0; j < 2; ++j) {
      const int i = t + 256 * j;        // 512 float4 total per matrix
      const int row = i >> 5;           // 32 float4 per row
      const int k = (i & 31) * 4;
      const int node = nodeBase + row;
      const float idg = invdeg[node];
      const float4 hs = *(const float4*)(h1 + (long)node * HDIM + k);
      const float4 ss = *(const float4*)(s2 + (long)node * HDIM + k);
      *(float4*)&Ah[row][k] = hs;
      float4 ms;
      ms.x = ss.x * idg; ms.y = ss.y * idg; ms.z = ss.z * idg; ms.w = ss.w * idg;
      *(float4*)&Am[row][k] = ms;
    }
  }
  __syncthreads();

  const int lane = threadIdx.x & 31;
  const int wave = threadIdx.x >> 5;
  const int colBase = wave * 16;
  const int col = colBase + (lane & 15);
  const int rowA = lane & 15;
  const int khalf = (lane >> 4) * 2;

  v8f c = {};
  #pragma unroll 4
  for (int kk = 0; kk < HDIM; kk += 4) {
    const int k0 = kk + khalf;
    v2f am = *(const v2f*)&Am[rowA][k0];
    v2f ah = *(const v2f*)&Ah[rowA][k0];
    v2f bl, br;
    bl.x = Wl[(long)k0 * HDIM + col]; bl.y = Wl[(long)(k0 + 1) * HDIM + col];
    br.x = Wr[(long)k0 * HDIM + col]; br.y = Wr[(long)(k0 + 1) * HDIM + col];
    c = __builtin_amdgcn_wmma_f32_16x16x4_f32(false, am, false, bl, (short)0, c, false, false);
    c = __builtin_amdgcn_wmma_f32_16x16x4_f32(false, ah, false, br, (short)0, c, false, false);
  }

  // fused: out[row] += sum_col (h2[row][col]) * W_lin[col]
  const float bias = b[col];
  const float wl = Wlin[col];
  float acc[8];
  #pragma unroll
  for (int v = 0; v < 8; ++v) acc[v] = (c[v] + bias) * wl;

  // reduce across the 16 lanes that share the same output row
  #pragma unroll
  for (int m = 1; m < 16; m <<= 1) {
    #pragma unroll
    for (int v = 0; v < 8; ++v) acc[v] += __shfl_xor(acc[v], m, 32);
  }

  const int rowOff = (lane >> 4) * 8;
  if ((lane & 15) == 0) {
    #pragma unroll
    for (int v = 0; v < 8; ++v)
      atomicAdd(&out[nodeBase + rowOff + v], acc[v]);
  }
}

// ---------------------------------------------------------------- launch
extern "C" void kernel_launch(void* const* d_in, const int* in_sizes, int n_in,
                              void* d_out, int out_size, void* d_ws, size_t ws_size,
                              hipStream_t stream) {
  const float* x     = (const float*)d_in[0];
  const int*   ei    = (const int*)d_in[1];
  const float* Wl1   = (const float*)d_in[2];
  const float* Wr1   = (const float*)d_in[3];
  const float* b1    = (const float*)d_in[4];
  const float* Wl2   = (const float*)d_in[5];
  const float* Wr2   = (const float*)d_in[6];
  const float* b2    = (const float*)d_in[7];
  const float* Wlin  = (const float*)d_in[8];
  const float* b_lin = (const float*)d_in[9];
  float* out = (float*)d_out;

  const long E = in_sizes[1] / 2;
  const int* src = ei;
  const int* dst = ei + E;

  // workspace layout (floats): deg[N] | s1[N*32] | h1[N*128] | s2[N*128]
  float* ws  = (float*)d_ws;
  float* deg = ws;
  float* s1  = deg + N_NODES;
  float* h1  = s1 + (long)N_NODES * F_IN;
  float* s2  = h1 + (long)N_NODES * HDIM;

  const int T = 256;

  // 1. zero deg+s1 (contiguous) and s2; init out = b_lin
  {
    long n0 = (long)N_NODES * (1 + F_IN);
    fill_zero_kernel<<<(int)((n0 + T - 1) / T), T, 0, stream>>>(deg, n0);
    long n1 = (long)N_NODES * HDIM;
    fill_zero_kernel<<<(int)((n1 + T - 1) / T), T, 0, stream>>>(s2, n1);
    init_out_kernel<<<(N_NODES + T - 1) / T, T, 0, stream>>>(out, b_lin, N_NODES);
  }

  // 2. degrees -> reciprocal
  degree_kernel<<<(int)((E + T - 1) / T), T, 0, stream>>>(dst, deg, E);
  invdeg_kernel<<<(N_NODES + T - 1) / T, T, 0, stream>>>(deg, N_NODES);

  // 3. scatter x (32-wide) into s1
  {
    long total = E * F_IN;
    scatter_add_kernel<<<(int)((total + T - 1) / T), T, 0, stream>>>(
        x, src, dst, s1, total, 5, 31);
  }

  // 4. layer 1 (WMMA f32 16x16x4): h1
  sage_layer1_kernel<<<N_NODES / 16, 256, 0, stream>>>(x, s1, deg, Wl1, Wr1, b1, h1);

  // 5. scatter h1 (128-wide) into s2
  {
    long total = E * HDIM;
    scatter_add_kernel<<<(int)((total + T - 1) / T), T, 0, stream>>>(
        h1, src, dst, s2, total, 7, 127);
  }

  // 6. layer 2 (WMMA) + fused final projection into out
  sage_layer2_kernel<<<N_NODES / 16, 256, 0, stream>>>(h1, s2, deg, Wl2, Wr2, b2,
                                                       Wlin, out);
}